// GNN_27934467293569
// MI455X (gfx1250) — compile-verified
//
#include <hip/hip_runtime.h>
#include <math.h>

// ---- problem constants (match reference) ----
#define NN  100000          // nodes
#define EE  1600000         // edges
#define DIM 64
#define EDIM 16
#define NEG_SLOPE 0.2f
#define BN_EPS 1e-5f
#define ET (EE + NN)        // edges incl. self loops

typedef float wf2 __attribute__((ext_vector_type(2)));
typedef float wf8 __attribute__((ext_vector_type(8)));

// ---- float atomic-max via order-preserving uint encoding ----
__device__ __forceinline__ unsigned f2ord(float f) {
    unsigned u = __float_as_uint(f);
    return (u & 0x80000000u) ? ~u : (u | 0x80000000u);
}
__device__ __forceinline__ float ord2f(unsigned u) {
    unsigned v = (u & 0x80000000u) ? (u & 0x7FFFFFFFu) : ~u;
    return __uint_as_float(v);
}

// ---------- precompute: degree + per-dst edge_attr sums ----------
__global__ void k_degloop(const long long* __restrict__ dst,
                          const float* __restrict__ ea,
                          float* __restrict__ deg, float* __restrict__ loop_sum) {
    int e = blockIdx.x * blockDim.x + threadIdx.x;
    if (e >= EE) return;
    int d = (int)dst[e];
    atomicAdd(deg + d, 1.0f);
    const float* a = ea + (size_t)e * EDIM;
    float* o = loop_sum + (size_t)d * EDIM;
#pragma unroll
    for (int c = 0; c < EDIM; ++c) atomicAdd(o + c, a[c]);
}

__global__ void k_loopnorm(float* __restrict__ loop_attr, const float* __restrict__ deg) {
    int i = blockIdx.x * blockDim.x + threadIdx.x;
    if (i >= NN * EDIM) return;
    loop_attr[i] /= fmaxf(deg[i >> 4], 1.0f);
}

// we_att[layer][r] = sum_c We[r*64+c] * att_e[c]   (a_edge = ea2 @ (We @ att_e))
__global__ void k_weatt(const float* __restrict__ We0, const float* __restrict__ ae0,
                        const float* __restrict__ We1, const float* __restrict__ ae1,
                        float* __restrict__ weatt) {
    int t = threadIdx.x;          // 32 threads
    int L = t >> 4, r = t & 15;
    const float* We = L ? We1 : We0;
    const float* ae = L ? ae1 : ae0;
    float s = 0.f;
#pragma unroll
    for (int c = 0; c < DIM; ++c) s += We[r * DIM + c] * ae[c];
    weatt[L * EDIM + r] = s;
}

// ---------- Y[nrows x 64] = X[nrows x 64] @ W[64 x 64] via V_WMMA_F32_16X16X4_F32 ----------
__global__ void k_gemm64(const float* __restrict__ X, const float* __restrict__ W,
                         float* __restrict__ Y, int nrows) {
    __shared__ float Ws[DIM * DIM];
    int tid = threadIdx.x;                       // 256 threads = 8 waves
    const float4* Wv = (const float4*)W;
    float4* Wsv = (float4*)Ws;
#pragma unroll
    for (int i = 0; i < 4; ++i) Wsv[tid + i * 256] = Wv[tid + i * 256];
    __syncthreads();

    int wave = tid >> 5, lane = tid & 31;
    int hi = lane >> 4, lo = lane & 15;
    int m0 = blockIdx.x * 128 + wave * 16;       // uniform per wave
    if (m0 >= nrows) return;                     // whole-wave exit, EXEC stays full

    int arow = m0 + lo;
    if (arow >= nrows) arow = nrows - 1;         // clamp (stores are guarded)
    // A tile: lane lo = row, half hi + vgpr = K; preload all 16 K-steps
    wf2 a[16];
#pragma unroll
    for (int kk = 0; kk < 16; ++kk)
        a[kk] = *(const wf2*)(X + (size_t)arow * DIM + kk * 4 + hi * 2);

    for (int nt = 0; nt < 4; ++nt) {
        int n0 = nt * 16;
        wf8 c = {};
#pragma unroll
        for (int kk = 0; kk < 16; ++kk) {
            wf2 b;
            b.x = Ws[(kk * 4 + 2 * hi + 0) * DIM + n0 + lo];
            b.y = Ws[(kk * 4 + 2 * hi + 1) * DIM + n0 + lo];
            c = __builtin_amdgcn_wmma_f32_16x16x4_f32(
                    false, a[kk], false, b, (short)0, c, false, false);
        }
#pragma unroll
        for (int r = 0; r < 8; ++r) {
            int row = m0 + r + 8 * hi;
            if (row < nrows) Y[(size_t)row * DIM + n0 + lo] = c[r];
        }
    }
}

// ---------- per-node attention scalars: a_src = xl . att_s, a_dst = xl . att_d ----------
__global__ void k_attscore(const float* __restrict__ xl, const float* __restrict__ as,
                           const float* __restrict__ ad,
                           float* __restrict__ a_src, float* __restrict__ a_dst) {
    int g = blockIdx.x * blockDim.x + threadIdx.x;
    int node = g >> 5, lane = threadIdx.x & 31;
    if (node >= NN) return;
    float2 v = *(const float2*)(xl + (size_t)node * DIM + lane * 2);
    float2 s = *(const float2*)(as + lane * 2);
    float2 d = *(const float2*)(ad + lane * 2);
    float ps = v.x * s.x + v.y * s.y;
    float pd = v.x * d.x + v.y * d.y;
#pragma unroll
    for (int off = 16; off; off >>= 1) {
        ps += __shfl_down(ps, off, 32);
        pd += __shfl_down(pd, off, 32);
    }
    if (lane == 0) { a_src[node] = ps; a_dst[node] = pd; }
}

// ---------- alpha + segment-max ----------
__global__ void k_alpha(const long long* __restrict__ src, const long long* __restrict__ dst,
                        const float* __restrict__ ea, const float* __restrict__ loop_attr,
                        const float* __restrict__ weatt,
                        const float* __restrict__ a_src, const float* __restrict__ a_dst,
                        float* __restrict__ alpha, unsigned* __restrict__ amax) {
    int e = blockIdx.x * blockDim.x + threadIdx.x;
    if (e >= ET) return;
    int s, d; const float* av;
    if (e < EE) { s = (int)src[e]; d = (int)dst[e]; av = ea + (size_t)e * EDIM; }
    else        { s = d = e - EE;  av = loop_attr + (size_t)s * EDIM; }
    float aedge = 0.f;
#pragma unroll
    for (int c = 0; c < EDIM; ++c) aedge += av[c] * weatt[c];
    float al = a_src[s] + a_dst[d] + aedge;
    al = (al >= 0.f) ? al : NEG_SLOPE * al;
    alpha[e] = al;
    atomicMax(amax + d, f2ord(al));
}

// ---------- exp + denominator ----------
__global__ void k_exp(const long long* __restrict__ dst, const float* __restrict__ alpha,
                      const unsigned* __restrict__ amax,
                      float* __restrict__ ex, float* __restrict__ den) {
    int e = blockIdx.x * blockDim.x + threadIdx.x;
    if (e >= ET) return;
    int d = (e < EE) ? (int)dst[e] : (e - EE);
    float v = expf(alpha[e] - ord2f(amax[d]));
    ex[e] = v;
    atomicAdd(den + d, v);
}

// ---------- scatter-accumulate: out[dst] += p * xl[src]  (one wave per edge) ----------
__global__ void k_acc(const long long* __restrict__ src, const long long* __restrict__ dst,
                      const float* __restrict__ xl, const float* __restrict__ ex,
                      const float* __restrict__ den, float* __restrict__ out) {
    long long T = (long long)blockIdx.x * blockDim.x + threadIdx.x;
    long long e = T >> 5;
    int lane = threadIdx.x & 31;
    if (e >= ET) return;
    int s, d;
    if (e < EE) { s = (int)src[e]; d = (int)dst[e]; }
    else        { s = d = (int)(e - EE); }
    float p = ex[e] / den[d];
    float2 v = *(const float2*)(xl + (size_t)s * DIM + lane * 2);
    float* o = out + (size_t)d * DIM + lane * 2;
    atomicAdd(o + 0, v.x * p);
    atomicAdd(o + 1, v.y * p);
}

// ---------- column sums / sumsq for batch-norm ----------
__global__ void k_colstats(const float* __restrict__ h,
                           float* __restrict__ col_sum, float* __restrict__ col_sq) {
    __shared__ float ssum[256], ssq[256];
    int col = threadIdx.x & 63;
    int rg = threadIdx.x >> 6;                   // 0..3
    float s = 0.f, q = 0.f;
    for (int row = blockIdx.x * 4 + rg; row < NN; row += gridDim.x * 4) {
        float v = h[(size_t)row * DIM + col];
        s += v; q += v * v;
    }
    ssum[threadIdx.x] = s; ssq[threadIdx.x] = q;
    __syncthreads();
    if (threadIdx.x < 64) {
        float ts = ssum[threadIdx.x] + ssum[threadIdx.x + 64] + ssum[threadIdx.x + 128] + ssum[threadIdx.x + 192];
        float tq = ssq[threadIdx.x]  + ssq[threadIdx.x + 64]  + ssq[threadIdx.x + 128]  + ssq[threadIdx.x + 192];
        atomicAdd(col_sum + threadIdx.x, ts);
        atomicAdd(col_sq + threadIdx.x, tq);
    }
}

// ---------- batch-norm (+ optional relu); conv bias cancels under BN ----------
__global__ void k_bn(const float* __restrict__ h, const float* __restrict__ col_sum,
                     const float* __restrict__ col_sq, const float* __restrict__ gamma,
                     const float* __restrict__ beta, float* __restrict__ out, int relu) {
    int i = blockIdx.x * blockDim.x + threadIdx.x;
    if (i >= NN * DIM) return;
    int c = i & 63;
    float m = col_sum[c] * (1.0f / NN);
    float v = col_sq[c] * (1.0f / NN) - m * m;
    float y = gamma[c] * (h[i] - m) * rsqrtf(v + BN_EPS) + beta[c];
    if (relu) y = fmaxf(y, 0.f);
    out[i] = y;
}

static inline int ceil_div(long long a, int b) { return (int)((a + b - 1) / b); }

extern "C" void kernel_launch(void* const* d_in, const int* in_sizes, int n_in,
                              void* d_out, int out_size, void* d_ws, size_t ws_size,
                              hipStream_t stream) {
    const float*     x    = (const float*)d_in[0];
    const long long* eidx = (const long long*)d_in[1];
    const long long* src  = eidx;
    const long long* dst  = eidx + EE;
    const float* ea   = (const float*)d_in[2];
    const float* W0   = (const float*)d_in[3];
    const float* We0  = (const float*)d_in[4];
    const float* as0  = (const float*)d_in[5];
    const float* ad0  = (const float*)d_in[6];
    const float* ae0  = (const float*)d_in[7];
    const float* g0   = (const float*)d_in[9];
    const float* be0  = (const float*)d_in[10];
    const float* W1   = (const float*)d_in[11];
    const float* We1  = (const float*)d_in[12];
    const float* as1  = (const float*)d_in[13];
    const float* ad1  = (const float*)d_in[14];
    const float* ae1  = (const float*)d_in[15];
    const float* gf   = (const float*)d_in[17];
    const float* bef  = (const float*)d_in[18];
    float* out = (float*)d_out;

    // ---- workspace carve-out (floats) ----
    float* ws = (float*)d_ws;
    size_t off = 0;
    float* xl    = ws + off; off += (size_t)NN * DIM;   // x @ W
    float* acc   = ws + off; off += (size_t)NN * DIM;   // gat accumulation
    float* h0    = ws + off; off += (size_t)NN * DIM;   // layer-0 output
    float* loopA = ws + off; off += (size_t)NN * EDIM;  // self-loop attrs
    float* deg   = ws + off; off += NN;
    float* asrc  = ws + off; off += NN;
    float* adst  = ws + off; off += NN;
    float* den   = ws + off; off += NN;
    unsigned* amax = (unsigned*)(ws + off); off += NN;
    float* alpha = ws + off; off += ET;
    float* exv   = ws + off; off += ET;
    float* weatt = ws + off; off += 32;
    float* csum  = ws + off; off += 64;
    float* csq   = ws + off; off += 64;

    const int B = 256;

    // ---- shared precompute (same for both layers) ----
    hipMemsetAsync(deg,   0, NN * sizeof(float), stream);
    hipMemsetAsync(loopA, 0, (size_t)NN * EDIM * sizeof(float), stream);
    k_degloop <<<ceil_div(EE, B), B, 0, stream>>>(dst, ea, deg, loopA);
    k_loopnorm<<<ceil_div((long long)NN * EDIM, B), B, 0, stream>>>(loopA, deg);
    k_weatt   <<<1, 32, 0, stream>>>(We0, ae0, We1, ae1, weatt);

    const float* Wl[2]  = {W0, W1};
    const float* asl[2] = {as0, as1};
    const float* adl[2] = {ad0, ad1};
    const float* gl[2]  = {g0, gf};
    const float* bl[2]  = {be0, bef};
    const float* xin[2] = {x, h0};
    float*       xo[2]  = {h0, out};

    for (int L = 0; L < 2; ++L) {
        hipMemsetAsync(amax, 0, NN * sizeof(unsigned), stream);
        hipMemsetAsync(den,  0, NN * sizeof(float), stream);
        hipMemsetAsync(acc,  0, (size_t)NN * DIM * sizeof(float), stream);
        hipMemsetAsync(csum, 0, 64 * sizeof(float), stream);
        hipMemsetAsync(csq,  0, 64 * sizeof(float), stream);

        k_gemm64  <<<ceil_div(NN, 128), B, 0, stream>>>(xin[L], Wl[L], xl, NN);
        k_attscore<<<ceil_div((long long)NN * 32, B), B, 0, stream>>>(xl, asl[L], adl[L], asrc, adst);
        k_alpha   <<<ceil_div(ET, B), B, 0, stream>>>(src, dst, ea, loopA, weatt + L * EDIM,
                                                      asrc, adst, alpha, amax);
        k_exp     <<<ceil_div(ET, B), B, 0, stream>>>(dst, alpha, amax, exv, den);
        k_acc     <<<ceil_div((long long)ET * 32, B), B, 0, stream>>>(src, dst, xl, exv, den, acc);
        k_colstats<<<512, B, 0, stream>>>(acc, csum, csq);
        k_bn      <<<ceil_div((long long)NN * DIM, B), B, 0, stream>>>(acc, csum, csq, gl[L], bl[L],
                                                                        xo[L], L == 0 ? 1 : 0);
    }
}